// MultiHeadGENLayer_41549513622181
// MI455X (gfx1250) — compile-verified
//
#include <hip/hip_runtime.h>
#include <hip/hip_bf16.h>
#include <math.h>

typedef float v2f __attribute__((ext_vector_type(2)));
typedef float v8f __attribute__((ext_vector_type(8)));

#define N_NODES 50000
#define N_EDGES 800000
#define DIM     128
#define HID     256

// ---------------------------------------------------------------------------
// Init workspace: mx = -inf, den = 0, num = 0, col stats = 0
// ---------------------------------------------------------------------------
__global__ void k_init(float* __restrict__ mx, float* __restrict__ den,
                       float* __restrict__ num, float* __restrict__ colstats) {
  long long total = (long long)N_NODES * DIM;
  long long stride = (long long)gridDim.x * blockDim.x;
  for (long long i = (long long)blockIdx.x * blockDim.x + threadIdx.x; i < total; i += stride) {
    mx[i]  = -INFINITY;
    den[i] = 0.0f;
    num[i] = 0.0f;
  }
  if (blockIdx.x == 0 && threadIdx.x < 2 * HID) colstats[threadIdx.x] = 0.0f;
}

// ---------------------------------------------------------------------------
// Edge pass 1: per-feature atomic max of s = (relu(x[src]+e)+eps)*t into mx[dst]
// Float max via sign-aware int-max / uint-min (mx initialized to -inf).
// ---------------------------------------------------------------------------
__global__ void k_edge_max(const float* __restrict__ x, const float* __restrict__ ea,
                           const float* __restrict__ ew, const float* __restrict__ eb,
                           const float* __restrict__ tptr, const int* __restrict__ ei,
                           float* __restrict__ mx) {
  const int e = blockIdx.x * 2 + (threadIdx.x >> 7);
  const int f = threadIdx.x & 127;
  const int src = ei[e];
  const int dst = ei[N_EDGES + e];
  const float t0 = tptr[0];
  const float enc = ea[2 * e] * ew[f] + ea[2 * e + 1] * ew[DIM + f] + eb[f];
  const float msg = fmaxf(x[src * DIM + f] + enc, 0.0f) + 1e-7f;
  const float s = msg * t0;
  float* addr = &mx[dst * DIM + f];
  if (s >= 0.0f)
    atomicMax((int*)addr, __float_as_int(s));
  else
    atomicMin((unsigned int*)addr, (unsigned int)__float_as_int(s));
}

// ---------------------------------------------------------------------------
// Edge pass 2: ex = exp(s - mx[dst]); den[dst] += ex; num[dst] += ex*msg
// ---------------------------------------------------------------------------
__global__ void k_edge_sum(const float* __restrict__ x, const float* __restrict__ ea,
                           const float* __restrict__ ew, const float* __restrict__ eb,
                           const float* __restrict__ tptr, const int* __restrict__ ei,
                           const float* __restrict__ mx,
                           float* __restrict__ den, float* __restrict__ num) {
  const int e = blockIdx.x * 2 + (threadIdx.x >> 7);
  const int f = threadIdx.x & 127;
  const int src = ei[e];
  const int dst = ei[N_EDGES + e];
  const float t0 = tptr[0];
  const float enc = ea[2 * e] * ew[f] + ea[2 * e + 1] * ew[DIM + f] + eb[f];
  const float msg = fmaxf(x[src * DIM + f] + enc, 0.0f) + 1e-7f;
  const float s = msg * t0;
  const float ex = expf(s - mx[dst * DIM + f]);
  atomicAdd(&den[dst * DIM + f], ex);
  atomicAdd(&num[dst * DIM + f], ex * msg);
}

// ---------------------------------------------------------------------------
// Node combine: h = num / max(den, 1e-16) + x   (written into mx's region)
// ---------------------------------------------------------------------------
__global__ void k_node_h(const float* __restrict__ x, const float* __restrict__ den,
                         const float* __restrict__ num, float* __restrict__ h) {
  const int i = blockIdx.x * blockDim.x + threadIdx.x; // grid covers N*DIM exactly
  h[i] = num[i] / fmaxf(den[i], 1e-16f) + x[i];
}

// ---------------------------------------------------------------------------
// GEMM1: h1 = h @ w1 + b1  (50000x128 @ 128x256), one 16x16 tile per wave,
// V_WMMA_F32_16X16X4_F32; also accumulates BN column sum / sumsq via atomics.
// ---------------------------------------------------------------------------
__global__ void k_gemm1(const float* __restrict__ h, const float* __restrict__ w1,
                        const float* __restrict__ b1, float* __restrict__ h1,
                        float* __restrict__ colsum, float* __restrict__ colsq) {
  const int wave = threadIdx.x >> 5;
  const int lane = threadIdx.x & 31;
  const int tile = blockIdx.x * 8 + wave;   // 50000 tiles = 3125 row-tiles x 16 col-tiles
  const int rbase = (tile >> 4) * 16;
  const int c0    = (tile & 15) * 16;
  const int m  = lane & 15;                 // A row / B col within tile
  const int kh = (lane >> 4) << 1;          // K sub-offset: 0 (lanes 0-15) or 2 (16-31)

  v8f acc = {0.f, 0.f, 0.f, 0.f, 0.f, 0.f, 0.f, 0.f};
  const float* arow = h + (rbase + m) * DIM;
  const float* bcol = w1 + (c0 + m);
  for (int k0 = 0; k0 < DIM; k0 += 4) {
    v2f a = *(const v2f*)(arow + k0 + kh);
    v2f b;
    b.x = bcol[(k0 + kh) * HID];
    b.y = bcol[(k0 + kh + 1) * HID];
    acc = __builtin_amdgcn_wmma_f32_16x16x4_f32(false, a, false, b, (short)0, acc,
                                                false, false);
  }
  const int col = c0 + m;
  const int rowoff = (lane >> 4) << 3;      // C/D: lanes 0-15 -> M=r, 16-31 -> M=r+8
  const float bb = b1[col];
  float psum = 0.0f, psq = 0.0f;
#pragma unroll
  for (int r = 0; r < 8; ++r) {
    const float v = acc[r] + bb;
    h1[(rbase + r + rowoff) * HID + col] = v;
    psum += v;
    psq  += v * v;
  }
  atomicAdd(&colsum[col], psum);
  atomicAdd(&colsq[col],  psq);
}

// ---------------------------------------------------------------------------
// Finalize BN stats: mu, inv_std per column (population variance)
// ---------------------------------------------------------------------------
__global__ void k_bnstats(const float* __restrict__ colsum, const float* __restrict__ colsq,
                          float* __restrict__ bmu, float* __restrict__ binv) {
  const int j = threadIdx.x;
  const float mu  = colsum[j] / (float)N_NODES;
  const float var = colsq[j] / (float)N_NODES - mu * mu;
  bmu[j]  = mu;
  binv[j] = rsqrtf(var + 1e-5f);
}

// ---------------------------------------------------------------------------
// GEMM2 + LayerNorm + ELU: each block owns 16 full rows.
// Stage BN+ReLU(h1) rows in LDS, 8 waves each compute one 16x16 tile of the
// 16x128 output block with WMMA, then block-local LN + ELU.
// ---------------------------------------------------------------------------
__global__ void k_gemm2(const float* __restrict__ h1, const float* __restrict__ w2,
                        const float* __restrict__ b2,
                        const float* __restrict__ bng, const float* __restrict__ bnb,
                        const float* __restrict__ bmu, const float* __restrict__ binv,
                        const float* __restrict__ lng, const float* __restrict__ lnb,
                        float* __restrict__ out) {
  __shared__ float As[16 * HID];   // 16 KB: BN+ReLU'd activations
  __shared__ float Ot[16 * DIM];   // 8 KB: pre-LN output block
  __shared__ float rmu[16], rsg[16];

  const int tid = threadIdx.x;
  const int rbase = blockIdx.x * 16;

  for (int i = tid; i < 16 * HID; i += 256) {
    const int c = i & (HID - 1);
    float v = h1[(rbase + (i >> 8)) * HID + c];
    v = (v - bmu[c]) * binv[c] * bng[c] + bnb[c];
    As[i] = fmaxf(v, 0.0f);
  }
  __syncthreads();

  const int wave = tid >> 5;
  const int lane = tid & 31;
  const int c0 = wave * 16;
  const int m  = lane & 15;
  const int kh = (lane >> 4) << 1;

  v8f acc = {0.f, 0.f, 0.f, 0.f, 0.f, 0.f, 0.f, 0.f};
  const float* bcol = w2 + (c0 + m);
  for (int k0 = 0; k0 < HID; k0 += 4) {
    v2f a = *(const v2f*)(&As[m * HID + k0 + kh]);
    v2f b;
    b.x = bcol[(k0 + kh) * DIM];
    b.y = bcol[(k0 + kh + 1) * DIM];
    acc = __builtin_amdgcn_wmma_f32_16x16x4_f32(false, a, false, b, (short)0, acc,
                                                false, false);
  }
  const int col = c0 + m;
  const int rowoff = (lane >> 4) << 3;
  const float bb = b2[col];
#pragma unroll
  for (int r = 0; r < 8; ++r) Ot[(r + rowoff) * DIM + col] = acc[r] + bb;
  __syncthreads();

  if (tid < 16) {
    float s = 0.0f, sq = 0.0f;
    for (int c = 0; c < DIM; ++c) {
      const float v = Ot[tid * DIM + c];
      s += v; sq += v * v;
    }
    const float mu = s / (float)DIM;
    rmu[tid] = mu;
    rsg[tid] = rsqrtf(sq / (float)DIM - mu * mu + 1e-5f);
  }
  __syncthreads();

  for (int i = tid; i < 16 * DIM; i += 256) {
    const int row = i >> 7;
    const int c   = i & 127;
    float v = (Ot[i] - rmu[row]) * rsg[row] * lng[c] + lnb[c];
    out[(rbase + row) * DIM + c] = v > 0.0f ? v : expf(v) - 1.0f;
  }
}

// ---------------------------------------------------------------------------
extern "C" void kernel_launch(void* const* d_in, const int* in_sizes, int n_in,
                              void* d_out, int out_size, void* d_ws, size_t ws_size,
                              hipStream_t stream) {
  const float* x        = (const float*)d_in[0];
  const float* ea       = (const float*)d_in[1];
  const float* ew       = (const float*)d_in[2];
  const float* eb       = (const float*)d_in[3];
  const float* t        = (const float*)d_in[4];
  const float* w1       = (const float*)d_in[5];
  const float* b1       = (const float*)d_in[6];
  const float* bn_gamma = (const float*)d_in[7];
  const float* bn_beta  = (const float*)d_in[8];
  const float* w2       = (const float*)d_in[9];
  const float* b2       = (const float*)d_in[10];
  const float* ln_gamma = (const float*)d_in[11];
  const float* ln_beta  = (const float*)d_in[12];
  const int*   ei       = (const int*)d_in[13];
  float* out = (float*)d_out;

  const long long ND = (long long)N_NODES * DIM;   // 6.4M
  const long long NH = (long long)N_NODES * HID;   // 12.8M
  float* ws     = (float*)d_ws;
  float* mx     = ws;                 // reused as h after edge passes
  float* den    = ws + ND;
  float* num    = ws + 2 * ND;
  float* h1     = ws + 3 * ND;
  float* colsum = ws + 3 * ND + NH;
  float* colsq  = colsum + HID;
  float* bmu    = colsq + HID;
  float* binv   = bmu + HID;
  float* hbuf   = mx;

  k_init<<<8192, 256, 0, stream>>>(mx, den, num, colsum);
  k_edge_max<<<N_EDGES / 2, 256, 0, stream>>>(x, ea, ew, eb, t, ei, mx);
  k_edge_sum<<<N_EDGES / 2, 256, 0, stream>>>(x, ea, ew, eb, t, ei, mx, den, num);
  k_node_h<<<(int)(ND / 256), 256, 0, stream>>>(x, den, num, hbuf);
  k_gemm1<<<6250, 256, 0, stream>>>(hbuf, w1, b1, h1, colsum, colsq);
  k_bnstats<<<1, 256, 0, stream>>>(colsum, colsq, bmu, binv);
  k_gemm2<<<3125, 256, 0, stream>>>(h1, w2, b2, bn_gamma, bn_beta, bmu, binv,
                                    ln_gamma, ln_beta, out);
}